// NetGRU_48859547959610
// MI455X (gfx1250) — compile-verified
//
#include <hip/hip_runtime.h>

// ---------------------------------------------------------------------------
// NetGRU on MI455X (gfx1250): fully fused hierarchical bi-GRU + attention.
//
// Layer 1: 8192 seqs [T=128, D=25] -> bi-GRU(H=16) -> attention -> Wm -> flow
// Layer 2:   64 seqs [N=128, M=16] -> bi-GRU(H=16) -> attention -> cv2 (out)
//
// Per block: 16 sequences, 2 waves (wave0 = forward scan, wave1 = backward).
// Gate GEMMs per step are v_wmma_f32_16x16x4_f32 (3 gate N-tiles of 16).
// Recurrence transpose goes through a tiny f32 LDS ping-pong (exact);
// the 128-step h history for attention is stored as bf16 (halves LDS ->
// 2 blocks/WGP -> all 4 SIMDs busy). Attention projection is also WMMA.
// ---------------------------------------------------------------------------

typedef __attribute__((ext_vector_type(2))) float v2f;
typedef __attribute__((ext_vector_type(8))) float v8f;

#define TSTEPS 128
#define GATES  3

__device__ __forceinline__ v8f wmma4(v2f a, v2f b, v8f c) {
    // D(16x16,f32) = A(16x4,f32) * B(4x16,f32) + C
    return __builtin_amdgcn_wmma_f32_16x16x4_f32(
        false, a, false, b, (short)0, c, false, false);
}

__device__ __forceinline__ float fast_rcp(float x) {
#if __has_builtin(__builtin_amdgcn_rcpf)
    return __builtin_amdgcn_rcpf(x);
#else
    return 1.0f / x;
#endif
}
__device__ __forceinline__ float sigmoidf(float x) {
    return fast_rcp(1.0f + __expf(-x));
}
__device__ __forceinline__ float fast_tanh(float x) {
#if __has_builtin(__builtin_amdgcn_tanhf)
    return __builtin_amdgcn_tanhf(x);
#else
    const float e = __expf(2.0f * x);
    return 1.0f - 2.0f * fast_rcp(e + 1.0f);
#endif
}

__device__ __forceinline__ unsigned short f2bf(float f) {
    return (unsigned short)(__float_as_uint(f) >> 16);
}
__device__ __forceinline__ float bf2f(unsigned short u) {
    return __uint_as_float(((unsigned)u) << 16);
}
__device__ __forceinline__ float bfpk_lo(unsigned u) { return __uint_as_float(u << 16); }
__device__ __forceinline__ float bfpk_hi(unsigned u) { return __uint_as_float(u & 0xffff0000u); }

// Branchless (no exec-mask) padded A-fragment load: clamp address, mask value.
template <int DIN>
__device__ __forceinline__ v2f load_ax(const float* __restrict__ xp, int kk, int hi) {
    const int k0 = kk * 4 + hi * 2;
    v2f r;
    if (((DIN + 3) / 4) * 4 == DIN) {
        r.x = xp[k0];
        r.y = xp[k0 + 1];
    } else {
        const int c0 = (k0     < DIN) ? k0     : (DIN - 1);
        const int c1 = (k0 + 1 < DIN) ? k0 + 1 : (DIN - 1);
        r.x = xp[c0] * ((k0     < DIN) ? 1.0f : 0.0f);
        r.y = xp[c1] * ((k0 + 1 < DIN) ? 1.0f : 0.0f);
    }
    return r;
}

// One direction of a GRU scan for 16 sequences, executed by ONE wave.
// hist: [t][seq(16)][dim(16)] bf16 history (for attention)
// pp:   [2][seq(16)][dim(16)] f32 ping-pong (exact recurrence transpose)
template <int DIN>
__device__ void gru_scan(const float* __restrict__ xbase, int seq_stride,
                         const float* __restrict__ Wih,   // [48, DIN]
                         const float* __restrict__ Whh,   // [48, 16]
                         const float* __restrict__ bih,   // [48]
                         const float* __restrict__ bhh,   // [48]
                         unsigned short* __restrict__ hist,
                         float* __restrict__ pp, bool reverse)
{
    constexpr int KX = (DIN + 3) / 4;
    const int lane = threadIdx.x & 31;
    const int row  = lane & 15;            // seq row (A) / out column (B)
    const int hi   = lane >> 4;            // half-wave: K pair select

    // --- Preload B fragments: B[k][n] = W[n][k] (W.T), 4x16 tiles ---
    v2f bx[GATES][KX];
    for (int g = 0; g < GATES; ++g)
        for (int kk = 0; kk < KX; ++kk) {
            const int n  = g * 16 + row;
            const int k0 = kk * 4 + hi * 2;
            v2f tmp;
            tmp.x = (k0     < DIN) ? Wih[n * DIN + k0]     : 0.0f;
            tmp.y = (k0 + 1 < DIN) ? Wih[n * DIN + k0 + 1] : 0.0f;
            bx[g][kk] = tmp;
        }
    v2f bh[GATES][4];
    for (int g = 0; g < GATES; ++g)
        for (int kk = 0; kk < 4; ++kk) {
            const int n  = g * 16 + row;
            const int k0 = kk * 4 + hi * 2;
            v2f tmp;
            tmp.x = Whh[n * 16 + k0];
            tmp.y = Whh[n * 16 + k0 + 1];
            bh[g][kk] = tmp;
        }

    // Combined biases (r,z gates); n gate keeps bhh inside r*(.)
    const float br  = bih[row]      + bhh[row];
    const float bz  = bih[16 + row] + bhh[16 + row];
    const float bxn = bih[32 + row];
    const float bhn = bhh[32 + row];

    v8f h = {};                                // h state in WMMA C-layout

    // Prefetch x A-fragments for step 0
    v2f axc[KX];
    {
        const int t0 = reverse ? (TSTEPS - 1) : 0;
        const float* xp = xbase + (size_t)row * seq_stride + (size_t)t0 * DIN;
        for (int kk = 0; kk < KX; ++kk) axc[kk] = load_ax<DIN>(xp, kk, hi);
    }

    for (int t = 0; t < TSTEPS; ++t) {
        const int tt = reverse ? (TSTEPS - 1 - t) : t;

        // Software-pipelined prefetch of next step's x fragments
        v2f axn[KX];
        if (t + 1 < TSTEPS) {
            const int tn = reverse ? (TSTEPS - 2 - t) : (t + 1);
            const float* xp = xbase + (size_t)row * seq_stride + (size_t)tn * DIN;
            for (int kk = 0; kk < KX; ++kk) axn[kk] = load_ax<DIN>(xp, kk, hi);
        } else {
            for (int kk = 0; kk < KX; ++kk) axn[kk] = axc[kk];
        }

        // gx = x_t @ Wih.T   (3 gate tiles)
        v8f gxr = {}, gxz = {}, gxn = {};
        for (int kk = 0; kk < KX; ++kk) {
            gxr = wmma4(axc[kk], bx[0][kk], gxr);
            gxz = wmma4(axc[kk], bx[1][kk], gxz);
            gxn = wmma4(axc[kk], bx[2][kk], gxn);
        }

        // gh = h_{t-1} @ Whh.T  (A transposed via f32 ping-pong in LDS)
        v8f ghr = {}, ghz = {}, ghn = {};
        if (t > 0) {
            const float* hp = pp + ((t - 1) & 1) * 256 + row * 16;
            v2f ah[4];
            for (int kk = 0; kk < 4; ++kk) {
                const int k0 = kk * 4 + hi * 2;
                v2f tmp;
                tmp.x = hp[k0];
                tmp.y = hp[k0 + 1];
                ah[kk] = tmp;
            }
            for (int kk = 0; kk < 4; ++kk) {
                ghr = wmma4(ah[kk], bh[0][kk], ghr);
                ghz = wmma4(ah[kk], bh[1][kk], ghz);
                ghn = wmma4(ah[kk], bh[2][kk], ghn);
            }
        }

        // GRU cell, elementwise in C-layout: element (seq = j + 8*hi, dim = row)
        float* ppw = pp + (t & 1) * 256;
        unsigned short* hw = hist + tt * 256;
        for (int j = 0; j < 8; ++j) {
            const float r  = sigmoidf(gxr[j] + ghr[j] + br);
            const float z  = sigmoidf(gxz[j] + ghz[j] + bz);
            const float n  = fast_tanh(gxn[j] + bxn + r * (ghn[j] + bhn));
            const float hv = (1.0f - z) * n + z * h[j];
            h[j] = hv;
            const int idx = (j + 8 * hi) * 16 + row;
            ppw[idx] = hv;            // exact value for next recurrence step
            hw[idx]  = f2bf(hv);      // bf16 history for attention
        }
        // No barrier needed: this wave is the only reader/writer of pp/hist.

        for (int kk = 0; kk < KX; ++kk) axc[kk] = axn[kk];
    }
}

// Attention pooling over TSTEPS for 16 sequences; whole block (2 waves).
// hf16/hb16: [t][seq][16] bf16. sc: [16][128] f32. cv: [16][32] f32.
__device__ void attention_cv(const unsigned short* __restrict__ hf16,
                             const unsigned short* __restrict__ hb16,
                             const float* __restrict__ Wa,   // [32,32]
                             const float* __restrict__ ba,   // [32]
                             const float* __restrict__ ctx,  // [32]
                             float* __restrict__ sc, float* __restrict__ cv)
{
    const int tid  = threadIdx.x;   // 0..63
    const int wv   = tid >> 5;
    const int lane = tid & 31;
    const int row  = lane & 15;
    const int hi   = lane >> 4;

    // B fragments for Wa.T (a = hcat @ Wa.T), K=32 -> 8 steps, N=32 -> 2 tiles
    v2f bwa[2][8];
    for (int nt = 0; nt < 2; ++nt)
        for (int kk = 0; kk < 8; ++kk) {
            const int n  = nt * 16 + row;
            const int k0 = kk * 4 + hi * 2;
            v2f tmp;
            tmp.x = Wa[n * 32 + k0];
            tmp.y = Wa[n * 32 + k0 + 1];
            bwa[nt][kk] = tmp;
        }
    const float ba0 = ba[row],  ba1 = ba[16 + row];
    const float cx0 = ctx[row], cx1 = ctx[16 + row];

    // Scores via WMMA: wave wv handles seqs {wv, wv+2, ...}; 8 t-tiles of 16.
    for (int si = 0; si < 8; ++si) {
        const int s = si * 2 + wv;
        for (int tb = 0; tb < 8; ++tb) {
            v8f c0 = {}, c1 = {};
            const int t = tb * 16 + row;          // A row = t within tile
            for (int kk = 0; kk < 8; ++kk) {
                const int k0 = (kk * 4 + hi * 2) & 15;
                const unsigned short* src = (kk < 4) ? hf16 : hb16;
                const unsigned u = *(const unsigned*)(src + t * 256 + s * 16 + k0);
                v2f a;
                a.x = bfpk_lo(u);
                a.y = bfpk_hi(u);
                c0 = wmma4(a, bwa[0][kk], c0);
                c1 = wmma4(a, bwa[1][kk], c1);
            }
            // score[t'] = sum_d tanh(a[t'][d]) * ctx[d]; reduce 16 lanes/half
            for (int j = 0; j < 8; ++j) {
                float v = fast_tanh(c0[j] + ba0) * cx0
                        + fast_tanh(c1[j] + ba1) * cx1;
                v += __shfl_xor(v, 1);
                v += __shfl_xor(v, 2);
                v += __shfl_xor(v, 4);
                v += __shfl_xor(v, 8);
                if (row == 0)
                    sc[s * 128 + tb * 16 + j + 8 * hi] = v;
            }
        }
    }
    __syncthreads();

    // Softmax over t + weighted sum: 4 lanes per sequence, 8 dims per lane.
    {
        const int s  = tid >> 2;
        const int ll = tid & 3;
        const int d0 = ll * 8;
        float m = -1e30f;
        for (int t = 0; t < 128; ++t) m = fmaxf(m, sc[s * 128 + t]);
        float denom = 0.0f;
        for (int t = 0; t < 128; ++t) denom += __expf(sc[s * 128 + t] - m);
        const float inv = fast_rcp(denom);
        float acc[8];
        for (int k = 0; k < 8; ++k) acc[k] = 0.0f;
        for (int t = 0; t < 128; ++t) {
            const float w = __expf(sc[s * 128 + t] - m) * inv;
            for (int k = 0; k < 8; ++k) {
                const int d = d0 + k;
                const unsigned short hv = (d < 16)
                    ? hf16[t * 256 + s * 16 + d]
                    : hb16[t * 256 + s * 16 + (d - 16)];
                acc[k] += w * bf2f(hv);
            }
        }
        for (int k = 0; k < 8; ++k) cv[s * 32 + d0 + k] = acc[k];
    }
    __syncthreads();
}

// LDS plan (bytes):
//   hf16 65536 | hb16 65536 | ppf 2048 | ppb 2048 | sc 8192 | cv 2048
//   total = 145408 B (~142 KB) -> 2 blocks per 320KB WGP
#define LDS_BYTES 145408

__global__ __launch_bounds__(64)
void gru_layer1_kernel(const float* __restrict__ x,
                       const float* __restrict__ Wih_f, const float* __restrict__ Whh_f,
                       const float* __restrict__ bih_f, const float* __restrict__ bhh_f,
                       const float* __restrict__ Wih_b, const float* __restrict__ Whh_b,
                       const float* __restrict__ bih_b, const float* __restrict__ bhh_b,
                       const float* __restrict__ Wa, const float* __restrict__ ba,
                       const float* __restrict__ ctx,
                       const float* __restrict__ Wm, const float* __restrict__ bm,
                       float* __restrict__ flow)
{
    extern __shared__ char lds_raw[];
    unsigned short* hf16 = (unsigned short*)(lds_raw);
    unsigned short* hb16 = (unsigned short*)(lds_raw + 65536);
    float* ppf = (float*)(lds_raw + 131072);
    float* ppb = (float*)(lds_raw + 133120);
    float* sc  = (float*)(lds_raw + 135168);
    float* cv  = (float*)(lds_raw + 143360);

    const int seq0 = blockIdx.x * 16;
    const float* xbase = x + (size_t)seq0 * 128 * 25;

    const bool rev = (threadIdx.x >> 5) != 0;   // wave-uniform
    gru_scan<25>(xbase, 128 * 25,
                 rev ? Wih_b : Wih_f, rev ? Whh_b : Whh_f,
                 rev ? bih_b : bih_f, rev ? bhh_b : bhh_f,
                 rev ? hb16 : hf16, rev ? ppb : ppf, rev);
    __syncthreads();

    attention_cv(hf16, hb16, Wa, ba, ctx, sc, cv);

    // flow = cv @ Wm.T + bm   -> [16 seq, 16]
    const int tid = threadIdx.x;
    for (int i = 0; i < 4; ++i) {
        const int idx = i * 64 + tid;   // 256 outputs
        const int s = idx >> 4;
        const int m = idx & 15;
        float acc = bm[m];
        for (int k = 0; k < 32; ++k) acc += cv[s * 32 + k] * Wm[m * 32 + k];
        flow[(size_t)(seq0 + s) * 16 + m] = acc;
    }
}

__global__ __launch_bounds__(64)
void gru_layer2_kernel(const float* __restrict__ flow,
                       const float* __restrict__ Wih_f, const float* __restrict__ Whh_f,
                       const float* __restrict__ bih_f, const float* __restrict__ bhh_f,
                       const float* __restrict__ Wih_b, const float* __restrict__ Whh_b,
                       const float* __restrict__ bih_b, const float* __restrict__ bhh_b,
                       const float* __restrict__ Wa, const float* __restrict__ ba,
                       const float* __restrict__ ctx,
                       float* __restrict__ out)
{
    extern __shared__ char lds_raw[];
    unsigned short* hf16 = (unsigned short*)(lds_raw);
    unsigned short* hb16 = (unsigned short*)(lds_raw + 65536);
    float* ppf = (float*)(lds_raw + 131072);
    float* ppb = (float*)(lds_raw + 133120);
    float* sc  = (float*)(lds_raw + 135168);
    float* cv  = (float*)(lds_raw + 143360);

    const int seq0 = blockIdx.x * 16;
    const float* xbase = flow + (size_t)seq0 * 128 * 16;

    const bool rev = (threadIdx.x >> 5) != 0;
    gru_scan<16>(xbase, 128 * 16,
                 rev ? Wih_b : Wih_f, rev ? Whh_b : Whh_f,
                 rev ? bih_b : bih_f, rev ? bhh_b : bhh_f,
                 rev ? hb16 : hf16, rev ? ppb : ppf, rev);
    __syncthreads();

    attention_cv(hf16, hb16, Wa, ba, ctx, sc, cv);

    // out = cv2  -> [16 seq, 32]
    const int tid = threadIdx.x;
    for (int i = 0; i < 8; ++i) {
        const int idx = i * 64 + tid;   // 512 outputs
        const int s = idx >> 5;
        const int k = idx & 31;
        out[(size_t)(seq0 + s) * 32 + k] = cv[s * 32 + k];
    }
}

extern "C" void kernel_launch(void* const* d_in, const int* in_sizes, int n_in,
                              void* d_out, int out_size, void* d_ws, size_t ws_size,
                              hipStream_t stream)
{
    (void)in_sizes; (void)n_in; (void)out_size; (void)ws_size;

    // setup_inputs() dict order
    const float* x     = (const float*)d_in[0];
    const float* Wih1f = (const float*)d_in[1];
    const float* Whh1f = (const float*)d_in[2];
    const float* bih1f = (const float*)d_in[3];
    const float* bhh1f = (const float*)d_in[4];
    const float* Wih1b = (const float*)d_in[5];
    const float* Whh1b = (const float*)d_in[6];
    const float* bih1b = (const float*)d_in[7];
    const float* bhh1b = (const float*)d_in[8];
    const float* Wih2f = (const float*)d_in[9];
    const float* Whh2f = (const float*)d_in[10];
    const float* bih2f = (const float*)d_in[11];
    const float* bhh2f = (const float*)d_in[12];
    const float* Wih2b = (const float*)d_in[13];
    const float* Whh2b = (const float*)d_in[14];
    const float* bih2b = (const float*)d_in[15];
    const float* bhh2b = (const float*)d_in[16];
    const float* Wa1   = (const float*)d_in[17];
    const float* ba1   = (const float*)d_in[18];
    const float* ctx1  = (const float*)d_in[19];
    const float* Wa2   = (const float*)d_in[20];
    const float* ba2   = (const float*)d_in[21];
    const float* ctx2  = (const float*)d_in[22];
    const float* Wm    = (const float*)d_in[23];
    const float* bm    = (const float*)d_in[24];

    float* flow = (float*)d_ws;           // [64*128, 16] = 512 KB
    float* out  = (float*)d_out;          // [64, 32]

    // Layer 1: 8192 sequences / 16 per block = 512 blocks, 2 waves each.
    gru_layer1_kernel<<<512, 64, LDS_BYTES, stream>>>(
        x, Wih1f, Whh1f, bih1f, bhh1f, Wih1b, Whh1b, bih1b, bhh1b,
        Wa1, ba1, ctx1, Wm, bm, flow);

    // Layer 2: 64 sequences / 16 per block = 4 blocks.
    gru_layer2_kernel<<<4, 64, LDS_BYTES, stream>>>(
        flow, Wih2f, Whh2f, bih2f, bhh2f, Wih2b, Whh2b, bih2b, bhh2b,
        Wa2, ba2, ctx2, out);
}